// Receiver_46076409152346
// MI455X (gfx1250) — compile-verified
//
#include <hip/hip_runtime.h>
#include <math.h>

typedef float v2f __attribute__((ext_vector_type(2)));
typedef float v8f __attribute__((ext_vector_type(8)));

#define B_   64
#define MSG_ 64
#define H_   512
#define D_   512
#define N_   2048

__device__ __forceinline__ float sigmf(float x) { return 1.0f / (1.0f + expf(-x)); }

// ---------------------------------------------------------------------------
// K0: wsum[o] = sum over first H columns of w_y1 row o   (w_y1 is [H, H+D])
// ---------------------------------------------------------------------------
__global__ void wsum_kernel(const float* __restrict__ w_y1, float* __restrict__ wsum) {
    int o = blockIdx.x * blockDim.x + threadIdx.x;   // 512 threads total
    const float* row = w_y1 + (size_t)o * (H_ + D_);
    float s = 0.f;
    for (int c = 0; c < H_; ++c) s += row[c];
    wsum[o] = s;
}

// ---------------------------------------------------------------------------
// K1: GRU cell -> new_state [B, H]
// ---------------------------------------------------------------------------
__global__ void gru_kernel(const float* __restrict__ msg, const float* __restrict__ state,
                           const float* __restrict__ w_ih, const float* __restrict__ w_hh,
                           const float* __restrict__ b_ih, const float* __restrict__ b_hh,
                           float* __restrict__ ns) {
    int idx = blockIdx.x * blockDim.x + threadIdx.x;  // 32768 threads
    int b = idx >> 9, j = idx & (H_ - 1);
    float ir = b_ih[j], iz = b_ih[H_ + j], inn = b_ih[2 * H_ + j];
    const float* wr = w_ih + (size_t)j * MSG_;
    const float* wz = w_ih + (size_t)(H_ + j) * MSG_;
    const float* wn = w_ih + (size_t)(2 * H_ + j) * MSG_;
    const float* xb = msg + (size_t)b * MSG_;
    for (int m = 0; m < MSG_; ++m) {
        float x = xb[m];
        ir += x * wr[m]; iz += x * wz[m]; inn += x * wn[m];
    }
    float hr = b_hh[j], hz = b_hh[H_ + j], hn = b_hh[2 * H_ + j];
    const float* vr = w_hh + (size_t)j * H_;
    const float* vz = w_hh + (size_t)(H_ + j) * H_;
    const float* vn = w_hh + (size_t)(2 * H_ + j) * H_;
    const float* hb = state + (size_t)b * H_;
    for (int k = 0; k < H_; ++k) {
        float h = hb[k];
        hr += h * vr[k]; hz += h * vz[k]; hn += h * vn[k];
    }
    float r = sigmf(ir + hr);
    float z = sigmf(iz + hz);
    float n = tanhf(inn + r * hn);
    ns[idx] = (1.0f - z) * n + z * hb[j];
}

// ---------------------------------------------------------------------------
// K2: stop head -> d_out[0:64] = round(sigmoid), d_out[64:128] = sigmoid
// ---------------------------------------------------------------------------
__global__ void stop_kernel(const float* __restrict__ ns, const float* __restrict__ w_stop,
                            const float* __restrict__ b_stop, float* __restrict__ out) {
    int b = threadIdx.x;                              // 64 threads
    const float* nb = ns + (size_t)b * H_;
    float s = b_stop[0];
    for (int k = 0; k < H_; ++k) s += nb[k] * w_stop[k];
    float sd = sigmf(s);
    out[64 + b] = sd;       // stop_dist
    out[b] = rintf(sd);     // stop_bit (round-half-even, matches jnp.round)
}

// ---------------------------------------------------------------------------
// K3: Cproj[n,o] = class_desc[n,:] . w_y1[o, H: ] + b_y1[o]   via f32 WMMA
//     A = cd [2048 x 512], B[k,o] = w_y1[o, 512+k], out [2048 x 512]
// ---------------------------------------------------------------------------
__global__ void cproj_wmma_kernel(const float* __restrict__ cd, const float* __restrict__ w_y1,
                                  const float* __restrict__ b_y1, float* __restrict__ cproj) {
    int wave = (blockIdx.x * blockDim.x + threadIdx.x) >> 5;  // 4096 waves
    int lane = threadIdx.x & 31;
    int tn = wave & 31;        // 32 tiles over o
    int tm = wave >> 5;        // 128 tiles over n
    int m = tm * 16 + (lane & 15);
    int o = tn * 16 + (lane & 15);
    int kh = (lane >> 4) * 2;
    const float* arow = cd + (size_t)m * D_;
    const float* brow = w_y1 + (size_t)o * (H_ + D_) + H_;
    v8f acc = {};
    for (int k0 = 0; k0 < D_; k0 += 4) {
        int k = k0 + kh;
        v2f a; a.x = arow[k]; a.y = arow[k + 1];
        v2f bb; bb.x = brow[k]; bb.y = brow[k + 1];
        acc = __builtin_amdgcn_wmma_f32_16x16x4_f32(false, a, false, bb,
                                                    (short)0, acc, false, false);
    }
    int mbase = tm * 16 + (lane >> 4) * 8;
    float bias = b_y1[o];
    for (int v = 0; v < 8; ++v)
        cproj[(size_t)(mbase + v) * H_ + o] = acc[v] + bias;
}

// ---------------------------------------------------------------------------
// K4: y[b,n] = b_y2 + sum_o w_y2[o] * relu(ns[b, n>>2]*wsum[o] + Cproj[n,o])
//     one wave per (b,n); writes y into d_out y-region
// ---------------------------------------------------------------------------
__global__ void score_kernel(const float* __restrict__ ns, const float* __restrict__ wsum,
                             const float* __restrict__ cproj, const float* __restrict__ w_y2,
                             const float* __restrict__ b_y2, float* __restrict__ y_out) {
    int gid = blockIdx.x * blockDim.x + threadIdx.x;  // 4,194,304 threads
    int wave = gid >> 5;                              // 131072 waves = B*N
    int lane = gid & 31;
    int b = wave >> 11;
    int n = wave & (N_ - 1);
    float v = ns[(size_t)b * H_ + (n >> 2)];
    const float* crow = cproj + (size_t)n * H_;
    float s = 0.f;
    for (int o = lane; o < H_; o += 32) {
        float t = fmaxf(v * wsum[o] + crow[o], 0.0f);
        s += w_y2[o] * t;
    }
    for (int off = 16; off; off >>= 1) s += __shfl_xor(s, off, 32);
    if (lane == 0) y_out[(size_t)b * N_ + n] = s + b_y2[0];
}

// ---------------------------------------------------------------------------
// K5: row softmax of y -> yscore [B, N]
// ---------------------------------------------------------------------------
__global__ void softmax_kernel(const float* __restrict__ y, float* __restrict__ yscore) {
    __shared__ float red[256];
    int row = blockIdx.x, tid = threadIdx.x;
    const float* yr = y + (size_t)row * N_;
    float m = -INFINITY;
    for (int n = tid; n < N_; n += 256) m = fmaxf(m, yr[n]);
    red[tid] = m; __syncthreads();
    for (int s = 128; s; s >>= 1) { if (tid < s) red[tid] = fmaxf(red[tid], red[tid + s]); __syncthreads(); }
    float rowmax = red[0]; __syncthreads();
    float* sr = yscore + (size_t)row * N_;
    float sum = 0.f;
    for (int n = tid; n < N_; n += 256) { float e = expf(yr[n] - rowmax); sr[n] = e; sum += e; }
    red[tid] = sum; __syncthreads();
    for (int s = 128; s; s >>= 1) { if (tid < s) red[tid] += red[tid + s]; __syncthreads(); }
    float inv = 1.0f / red[0];
    for (int n = tid; n < N_; n += 256) sr[n] *= inv;
}

// ---------------------------------------------------------------------------
// K6: dinf = yscore @ class_desc  ([64,2048] x [2048,512]) via f32 WMMA
// ---------------------------------------------------------------------------
__global__ void dinf_wmma_kernel(const float* __restrict__ yscore, const float* __restrict__ cd,
                                 float* __restrict__ dinf) {
    int wave = (blockIdx.x * blockDim.x + threadIdx.x) >> 5;  // 128 waves
    int lane = threadIdx.x & 31;
    int tn = wave & 31;   // 32 tiles over d
    int tm = wave >> 5;   // 4 tiles over b
    int m = tm * 16 + (lane & 15);
    int o = tn * 16 + (lane & 15);
    int kh = (lane >> 4) * 2;
    const float* arow = yscore + (size_t)m * N_;
    v8f acc = {};
    for (int k0 = 0; k0 < N_; k0 += 4) {
        int k = k0 + kh;
        v2f a; a.x = arow[k]; a.y = arow[k + 1];
        v2f bb; bb.x = cd[(size_t)k * D_ + o]; bb.y = cd[(size_t)(k + 1) * D_ + o];
        acc = __builtin_amdgcn_wmma_f32_16x16x4_f32(false, a, false, bb,
                                                    (short)0, acc, false, false);
    }
    int mbase = tm * 16 + (lane >> 4) * 8;
    for (int v = 0; v < 8; ++v)
        dinf[(size_t)(mbase + v) * D_ + o] = acc[v];
}

// ---------------------------------------------------------------------------
// K7: mstate = tanh(ns @ w_state.T + b_state + dinf @ w_desc.T)
// ---------------------------------------------------------------------------
__global__ void mstate_kernel(const float* __restrict__ ns, const float* __restrict__ dinf,
                              const float* __restrict__ w_state, const float* __restrict__ b_state,
                              const float* __restrict__ w_desc, float* __restrict__ mstate) {
    int idx = blockIdx.x * blockDim.x + threadIdx.x;  // 32768 threads
    int b = idx >> 9, o = idx & (H_ - 1);
    const float* nb = ns + (size_t)b * H_;
    const float* db = dinf + (size_t)b * D_;
    const float* ws = w_state + (size_t)o * H_;
    const float* wd = w_desc + (size_t)o * D_;
    float s = b_state[o];
    for (int k = 0; k < H_; ++k) s += nb[k] * ws[k];
    for (int k = 0; k < D_; ++k) s += db[k] * wd[k];
    mstate[idx] = tanhf(s);
}

// ---------------------------------------------------------------------------
// K8: message_dist = sigmoid(mstate @ w_msg.T + b_msg); outp_msg = round(.)
// ---------------------------------------------------------------------------
__global__ void msg_kernel(const float* __restrict__ mstate, const float* __restrict__ w_msg,
                           const float* __restrict__ b_msg, float* __restrict__ out) {
    int idx = blockIdx.x * blockDim.x + threadIdx.x;  // 4096 threads
    int b = idx >> 6, m = idx & (MSG_ - 1);
    const float* mb = mstate + (size_t)b * H_;
    const float* wm = w_msg + (size_t)m * H_;
    float s = b_msg[m];
    for (int k = 0; k < H_; ++k) s += mb[k] * wm[k];
    float md = sigmf(s);
    out[4224 + idx] = md;          // message_dist region
    out[128 + idx] = rintf(md);    // outp_msg region
}

// ---------------------------------------------------------------------------
extern "C" void kernel_launch(void* const* d_in, const int* in_sizes, int n_in,
                              void* d_out, int out_size, void* d_ws, size_t ws_size,
                              hipStream_t stream) {
    const float* message = (const float*)d_in[0];
    const float* state   = (const float*)d_in[1];
    // d_in[2] = image (unused by the reference)
    const float* cd      = (const float*)d_in[3];
    const float* w_ih    = (const float*)d_in[4];
    const float* w_hh    = (const float*)d_in[5];
    const float* b_ih    = (const float*)d_in[6];
    const float* b_hh    = (const float*)d_in[7];
    const float* w_stop  = (const float*)d_in[8];
    const float* b_stop  = (const float*)d_in[9];
    const float* w_y1    = (const float*)d_in[10];
    const float* b_y1    = (const float*)d_in[11];
    const float* w_y2    = (const float*)d_in[12];
    const float* b_y2    = (const float*)d_in[13];
    const float* w_state = (const float*)d_in[14];
    const float* b_state = (const float*)d_in[15];
    const float* w_desc  = (const float*)d_in[16];
    const float* w_msg   = (const float*)d_in[17];
    const float* b_msg   = (const float*)d_in[18];

    float* out = (float*)d_out;
    // d_out layout: stop_bit[64] | stop_dist[64] | outp_msg[4096] |
    //               message_dist[4096] | y[64*2048]
    float* y_out = out + 8320;

    float* w = (float*)d_ws;
    float* ns     = w;                 // 32768
    float* wsum   = w + 32768;         // 512
    float* cproj  = w + 33280;         // 2048*512
    float* yscore = w + 1081856;       // 64*2048
    float* dinf   = w + 1212928;       // 64*512
    float* mstate = w + 1245696;       // 64*512   (total ~5.1 MB)

    wsum_kernel      <<<2,    256, 0, stream>>>(w_y1, wsum);
    gru_kernel       <<<128,  256, 0, stream>>>(message, state, w_ih, w_hh, b_ih, b_hh, ns);
    stop_kernel      <<<1,     64, 0, stream>>>(ns, w_stop, b_stop, out);
    cproj_wmma_kernel<<<1024, 128, 0, stream>>>(cd, w_y1, b_y1, cproj);
    score_kernel     <<<16384,256, 0, stream>>>(ns, wsum, cproj, w_y2, b_y2, y_out);
    softmax_kernel   <<<64,   256, 0, stream>>>(y_out, yscore);
    dinf_wmma_kernel <<<32,   128, 0, stream>>>(yscore, cd, dinf);
    mstate_kernel    <<<128,  256, 0, stream>>>(ns, dinf, w_state, b_state, w_desc, mstate);
    msg_kernel       <<<16,   256, 0, stream>>>(mstate, w_msg, b_msg, out);
}